// KnowledgeModule_70952859730514
// MI455X (gfx1250) — compile-verified
//
#include <hip/hip_runtime.h>
#include <cstdint>

#define NVARS   524288
#define WIDTH_  524288
#define FANIN_  8
#define NEDGES  (WIDTH_ * FANIN_)       // 4,194,304 edges per layer
#define EPS_    1e-15f
#define CUTOFF_ (-0.69314718055994530942f)  // -log(2)

// ---------------------------------------------------------------------------
// encode_input: enc = [-inf, 0, w0, log1mexp(w0), w1, log1mexp(w1), ...]
// Grid exactly covers NVARS (2048 x 256), no bounds check needed.
// ---------------------------------------------------------------------------
__global__ __launch_bounds__(256) void encode_kernel(const float* __restrict__ w,
                                                     float* __restrict__ enc) {
    const int i = blockIdx.x * 256 + threadIdx.x;
    if (i == 0) { enc[0] = -__builtin_inff(); enc[1] = 0.0f; }
    const float x = w[i];   // x = log(u) < 0
    const float neg = (x > CUTOFF_) ? __logf(-expm1f(x))
                                    : log1pf(-__expf(x));
    float2 o; o.x = x; o.y = neg;
    ((float2*)(enc + 2))[i] = o;    // interleaved [w_i, log1mexp(w_i)]
}

// ---------------------------------------------------------------------------
// One circuit layer: node j reduces gathered x[ptr[8j..8j+7]].
//   LOGSUM = false : log-product  (8-way sum)
//   LOGSUM = true  : stabilized logsumexp with nan_to_num(g - m, nan=0)
//
// The 32 B/lane pointer slice is staged LDS-ward with the CDNA5 async path
// (GLOBAL_LOAD_ASYNC_TO_LDS_B128, ASYNCcnt) using a non-temporal hint so the
// use-once 128 MB index stream does not evict the L2-resident gather table.
// Each lane only reads back the bytes its own async op wrote, so
// s_wait_asynccnt 0 is the only required synchronization.
// ---------------------------------------------------------------------------
template <bool LOGSUM>
__global__ __launch_bounds__(256) void layer_kernel(const float* __restrict__ x,
                                                    const int*   __restrict__ ptr,
                                                    float*       __restrict__ out) {
    __shared__ __align__(16) int sp[256 * FANIN_];   // 8 KB / block
    const int tid = threadIdx.x;
    const int j   = blockIdx.x * 256 + tid;          // node id (grid == WIDTH_)

    // LDS byte address (low 32 bits of the flat shared address == AS3 offset)
    const uint32_t lds_off = (uint32_t)(uintptr_t)(&sp[tid * FANIN_]);
    const uint32_t g_off   = (uint32_t)j * (FANIN_ * 4u);  // byte offset into ptr

    // Async copy 32 B of pointers: offset applies to both LDS and global addr.
    asm volatile(
        "global_load_async_to_lds_b128 %0, %1, %2 th:TH_LOAD_NT\n\t"
        "global_load_async_to_lds_b128 %0, %1, %2 offset:16 th:TH_LOAD_NT"
        :
        : "v"(lds_off), "v"(g_off), "s"(ptr)
        : "memory");
    asm volatile("s_wait_asynccnt 0" ::: "memory");

    const int4* sp4 = (const int4*)(&sp[tid * FANIN_]);
    const int4 a = sp4[0];
    const int4 b = sp4[1];

    // Random 4 B gathers; table is 2-4 MB -> resident in 192 MB L2.
    const float g0 = x[a.x], g1 = x[a.y], g2 = x[a.z], g3 = x[a.w];
    const float g4 = x[b.x], g5 = x[b.y], g6 = x[b.z], g7 = x[b.w];

    float r;
    if (!LOGSUM) {
        r = ((g0 + g1) + (g2 + g3)) + ((g4 + g5) + (g6 + g7));
    } else {
        const float m = fmaxf(fmaxf(fmaxf(g0, g1), fmaxf(g2, g3)),
                              fmaxf(fmaxf(g4, g5), fmaxf(g6, g7)));
        float s = 0.0f;
#define ACC_(g) { float y = (g) - m; y = (y != y) ? 0.0f : y; s += __expf(y); }
        ACC_(g0) ACC_(g1) ACC_(g2) ACC_(g3)
        ACC_(g4) ACC_(g5) ACC_(g6) ACC_(g7)
#undef ACC_
        r = __logf(s + EPS_) + m;   // m = -inf (all-dead segment) -> r = -inf
    }
    out[j] = r;
}

// ---------------------------------------------------------------------------
// Driver: encode, then 8 dependent layer launches (stream order = global sync).
// d_ws layout: [enc: 1,048,580 f32][bufA: 524,288 f32][bufB: 524,288 f32]
// ---------------------------------------------------------------------------
extern "C" void kernel_launch(void* const* d_in, const int* in_sizes, int n_in,
                              void* d_out, int out_size, void* d_ws, size_t ws_size,
                              hipStream_t stream) {
    (void)in_sizes; (void)n_in; (void)out_size; (void)ws_size;

    const float* w    = (const float*)d_in[0];   // weights [NVARS]
    const int*   ptrs = (const int*)d_in[1];     // ptrs [8, NEDGES]
    // d_in[2] (csrs) is repeat(arange(WIDTH), 8): implicit, never read.

    float* enc  = (float*)d_ws;                       // 2 + 2*NVARS floats
    float* bufA = enc + (2 + 2 * (size_t)NVARS) + 2;  // +2 pad -> 16 B aligned
    float* bufB = bufA + WIDTH_;

    const dim3 blk(256);
    const dim3 grdV(NVARS / 256);    // 2048
    const dim3 grdW(WIDTH_ / 256);   // 2048

    encode_kernel<<<grdV, blk, 0, stream>>>(w, enc);

    const size_t E = (size_t)NEDGES;
    layer_kernel<false><<<grdW, blk, 0, stream>>>(enc,  ptrs + 0 * E, bufA);
    layer_kernel<true ><<<grdW, blk, 0, stream>>>(bufA, ptrs + 1 * E, bufB);
    layer_kernel<false><<<grdW, blk, 0, stream>>>(bufB, ptrs + 2 * E, bufA);
    layer_kernel<true ><<<grdW, blk, 0, stream>>>(bufA, ptrs + 3 * E, bufB);
    layer_kernel<false><<<grdW, blk, 0, stream>>>(bufB, ptrs + 4 * E, bufA);
    layer_kernel<true ><<<grdW, blk, 0, stream>>>(bufA, ptrs + 5 * E, bufB);
    layer_kernel<false><<<grdW, blk, 0, stream>>>(bufB, ptrs + 6 * E, bufA);
    layer_kernel<true ><<<grdW, blk, 0, stream>>>(bufA, ptrs + 7 * E, (float*)d_out);
}